// MultiHeadAttentionLayer_10196252360941
// MI455X (gfx1250) — compile-verified
//
#include <hip/hip_runtime.h>
#include <hip/hip_bf16.h>
#include <math.h>

#define FDIM  64
#define NHEAD 8
#define HDIM  8
#define CLIPV 5.0f

typedef __attribute__((ext_vector_type(2))) float v2f;
typedef __attribute__((ext_vector_type(4))) float v4f;
typedef __attribute__((ext_vector_type(8))) float v8f;

__device__ __forceinline__ float clipf(float x) {
    return fminf(fmaxf(x, -CLIPV), CLIPV);
}

// ---------------------------------------------------------------------------
// Kernel 0: zero the accumulators (wV + z) in workspace
// ---------------------------------------------------------------------------
__global__ void zero_kernel(float* __restrict__ p, long long n) {
    long long i = (long long)blockIdx.x * blockDim.x + threadIdx.x;
    long long stride = (long long)gridDim.x * blockDim.x;
    for (; i < n; i += stride) p[i] = 0.0f;
}

// ---------------------------------------------------------------------------
// Kernel 1: node projections Q/K/V = X @ W^T + b  via V_WMMA_F32_16X16X4_F32
// One wave computes a 16-row x 64-col output tile. A-fragment layout per ISA
// 7.12.2 (32-bit A 16x4): VGPR0 = K={0|2}, VGPR1 = K={1|3} across half-waves.
// X is streamed once -> NT loads; Q/K/V are re-read by the edge kernel ->
// default (RT) stores so they stay resident in the 192MB L2.
// ---------------------------------------------------------------------------
__global__ void __launch_bounds__(256)
node_proj_kernel(const float* __restrict__ X,
                 const float* __restrict__ Wq, const float* __restrict__ bq,
                 const float* __restrict__ Wk, const float* __restrict__ bk,
                 const float* __restrict__ Wv, const float* __restrict__ bv,
                 float* __restrict__ Q, float* __restrict__ K, float* __restrict__ V,
                 int nrows)
{
    __shared__ __attribute__((aligned(16))) float sW[FDIM * FDIM];
    __shared__ __attribute__((aligned(16))) float sB[FDIM];

    const int tid  = threadIdx.x;
    const int wave = tid >> 5;
    const int lane = tid & 31;
    const int lo   = lane & 15;
    const int hi   = lane >> 4;
    const int tileRow = (blockIdx.x * 8 + wave) * 16;

    // Preload the 16x64 X tile as A fragments (reused for all 3 matrices)
    float a_reg[32];
    if (tileRow < nrows) {
        int r = tileRow + lo; if (r >= nrows) r = nrows - 1;
        const float* xrow = X + (long long)r * FDIM + 2 * hi;
        #pragma unroll
        for (int kc = 0; kc < 16; ++kc) {
            v2f a = __builtin_nontemporal_load((const v2f*)(xrow + 4 * kc));
            a_reg[2 * kc]     = a.x;
            a_reg[2 * kc + 1] = a.y;
        }
    }

    const float* Ws[3] = {Wq, Wk, Wv};
    const float* Bs[3] = {bq, bk, bv};
    float*       Os[3] = {Q, K, V};

    #pragma unroll
    for (int m = 0; m < 3; ++m) {
        __syncthreads();
        #pragma unroll
        for (int i = 0; i < (FDIM * FDIM) / 256; ++i)
            sW[tid + 256 * i] = Ws[m][tid + 256 * i];
        if (tid < FDIM) sB[tid] = Bs[m][tid];
        __syncthreads();

        if (tileRow >= nrows) continue;
        float* outp = Os[m];

        #pragma unroll
        for (int t = 0; t < 4; ++t) {
            v8f acc = {0.f, 0.f, 0.f, 0.f, 0.f, 0.f, 0.f, 0.f};
            const float* wrow = sW + (16 * t + lo) * FDIM + 2 * hi;
            #pragma unroll
            for (int kc = 0; kc < 16; ++kc) {
                v2f a; a.x = a_reg[2 * kc]; a.y = a_reg[2 * kc + 1];
                v2f b = *(const v2f*)(wrow + 4 * kc);
                acc = __builtin_amdgcn_wmma_f32_16x16x4_f32(
                        false, a, false, b, (short)0, acc, false, false);
            }
            float bias = sB[16 * t + lo];
            if (tileRow + 16 <= nrows) {
                // full tile: wave-uniform fast path, unguarded coalesced stores
                #pragma unroll
                for (int r = 0; r < 8; ++r)
                    outp[(long long)(tileRow + r + 8 * hi) * FDIM + 16 * t + lo]
                        = acc[r] + bias;
            } else {
                #pragma unroll
                for (int r = 0; r < 8; ++r) {
                    int row = tileRow + r + 8 * hi;
                    if (row < nrows)
                        outp[(long long)row * FDIM + 16 * t + lo] = acc[r] + bias;
                }
            }
        }
    }
}

// ---------------------------------------------------------------------------
// Kernel 2: fused edge kernel.
//  Phase 1: proj_e (16 edges x 64) via WMMA -> LDS stage per wave.
//  Phase 2: per (edge,head) pair: score = clip(dot(K[src],Q[dst])/sqrt(8)),
//           e_out = score * proj_e, w = exp(clip(score * sum_d proj_e)),
//           atomic scatter of w*V[src] and w by dst.
//  edge_feats reads and e_out writes are NT (streamed once, 512MB combined)
//  so the L2 stays dedicated to the Q/K/V gather tables and wV/z atomics.
// ---------------------------------------------------------------------------
__global__ void __launch_bounds__(256)
edge_kernel(const float* __restrict__ EF,
            const int* __restrict__ src, const int* __restrict__ dst,
            const float* __restrict__ We, const float* __restrict__ be,
            const float* __restrict__ Q, const float* __restrict__ K,
            const float* __restrict__ V,
            float* __restrict__ e_out, float* __restrict__ wV,
            float* __restrict__ z, int nedges)
{
    __shared__ __attribute__((aligned(16))) float sW[FDIM * FDIM];
    __shared__ __attribute__((aligned(16))) float sB[FDIM];
    __shared__ __attribute__((aligned(16))) float sPE[8][16 * FDIM];

    const int tid  = threadIdx.x;
    const int wave = tid >> 5;
    const int lane = tid & 31;
    const int lo   = lane & 15;
    const int hi   = lane >> 4;
    const int eb   = (blockIdx.x * 8 + wave) * 16;

    #pragma unroll
    for (int i = 0; i < (FDIM * FDIM) / 256; ++i)
        sW[tid + 256 * i] = We[tid + 256 * i];
    if (tid < FDIM) sB[tid] = be[tid];
    __syncthreads();

    // ---- Phase 1: proj_e for this wave's 16 edges ----
    if (eb < nedges) {
        int r = eb + lo; if (r >= nedges) r = nedges - 1;
        const float* xrow = EF + (long long)r * FDIM + 2 * hi;
        // speculative prefetch of the next block's tile (same wave slot)
        __builtin_prefetch(xrow + (long long)8 * 16 * FDIM, 0, 0);
        float a_reg[32];
        #pragma unroll
        for (int kc = 0; kc < 16; ++kc) {
            v2f a = __builtin_nontemporal_load((const v2f*)(xrow + 4 * kc));
            a_reg[2 * kc]     = a.x;
            a_reg[2 * kc + 1] = a.y;
        }
        #pragma unroll
        for (int t = 0; t < 4; ++t) {
            v8f acc = {0.f, 0.f, 0.f, 0.f, 0.f, 0.f, 0.f, 0.f};
            const float* wrow = sW + (16 * t + lo) * FDIM + 2 * hi;
            #pragma unroll
            for (int kc = 0; kc < 16; ++kc) {
                v2f a; a.x = a_reg[2 * kc]; a.y = a_reg[2 * kc + 1];
                v2f b = *(const v2f*)(wrow + 4 * kc);
                acc = __builtin_amdgcn_wmma_f32_16x16x4_f32(
                        false, a, false, b, (short)0, acc, false, false);
            }
            float bias = sB[16 * t + lo];
            #pragma unroll
            for (int r8 = 0; r8 < 8; ++r8)
                sPE[wave][(r8 + 8 * hi) * FDIM + 16 * t + lo] = acc[r8] + bias;
        }
    }
    __syncthreads();

    // ---- Phase 2: per-(edge,head) attention + scatter ----
    if (eb < nedges) {
        const float rsq = 0.35355339059327373f; // 1/sqrt(D)
        const bool fullTile = (eb + 16 <= nedges);
        #pragma unroll
        for (int it = 0; it < 4; ++it) {
            int p  = it * 32 + lane;
            int el = p >> 3;        // local edge 0..15
            int h  = p & 7;         // head
            int e  = eb + el;
            if (!fullTile && e >= nedges) continue;

            int si = src[e];
            int di = dst[e];

            const float4* Kp = (const float4*)(K + (long long)si * FDIM + h * HDIM);
            const float4* Qp = (const float4*)(Q + (long long)di * FDIM + h * HDIM);
            float4 k0 = Kp[0], k1 = Kp[1];
            float4 q0 = Qp[0], q1 = Qp[1];
            float dot = k0.x * q0.x + k0.y * q0.y + k0.z * q0.z + k0.w * q0.w
                      + k1.x * q1.x + k1.y * q1.y + k1.z * q1.z + k1.w * q1.w;
            float s = clipf(dot * rsq);

            const float4* pe4 = (const float4*)(&sPE[wave][el * FDIM + h * HDIM]);
            float4 p0 = pe4[0], p1 = pe4[1];
            float spe = p0.x + p0.y + p0.z + p0.w + p1.x + p1.y + p1.z + p1.w;
            float wgt = __expf(clipf(s * spe));

            // e_out = s * proj_e : streamed-once -> non-temporal 16B stores
            v4f o0 = {s * p0.x, s * p0.y, s * p0.z, s * p0.w};
            v4f o1 = {s * p1.x, s * p1.y, s * p1.z, s * p1.w};
            v4f* eo = (v4f*)(e_out + (long long)e * FDIM + h * HDIM);
            __builtin_nontemporal_store(o0, eo);
            __builtin_nontemporal_store(o1, eo + 1);

            // segment-sum scatter: wV[dst] += w * V[src],  z[dst] += w
            const float* Vp  = V  + (long long)si * FDIM + h * HDIM;
            float*       wVp = wV + (long long)di * FDIM + h * HDIM;
            #pragma unroll
            for (int d = 0; d < 8; ++d)
                atomicAdd(&wVp[d], wgt * Vp[d]);
            atomicAdd(&z[(long long)di * NHEAD + h], wgt);
        }
    }
}

// ---------------------------------------------------------------------------
// Kernel 3: h_out = wV / (z + 1e-6)   (h_out write-once -> NT store)
// ---------------------------------------------------------------------------
__global__ void normalize_kernel(const float* __restrict__ wV,
                                 const float* __restrict__ z,
                                 float* __restrict__ h_out, int total)
{
    int i = blockIdx.x * blockDim.x + threadIdx.x;
    if (i < total) {
        int node = i >> 6;          // / FDIM
        int h    = (i >> 3) & 7;    // head
        float v = wV[i] / (z[node * NHEAD + h] + 1e-6f);
        __builtin_nontemporal_store(v, h_out + i);
    }
}

// ---------------------------------------------------------------------------
extern "C" void kernel_launch(void* const* d_in, const int* in_sizes, int n_in,
                              void* d_out, int out_size, void* d_ws, size_t ws_size,
                              hipStream_t stream)
{
    const float* node_feats = (const float*)d_in[0];
    const float* edge_feats = (const float*)d_in[1];
    const int*   src        = (const int*)d_in[2];
    const int*   dst        = (const int*)d_in[3];
    const float* Wq = (const float*)d_in[4];
    const float* bq = (const float*)d_in[5];
    const float* Wk = (const float*)d_in[6];
    const float* bk = (const float*)d_in[7];
    const float* Wv = (const float*)d_in[8];
    const float* bv = (const float*)d_in[9];
    const float* We = (const float*)d_in[10];
    const float* be = (const float*)d_in[11];

    const int n = in_sizes[0] / FDIM;   // N nodes
    const int e = in_sizes[1] / FDIM;   // E edges

    // Workspace layout: Q | K | V | wV (each N*64) | z (N*8)
    float* ws = (float*)d_ws;
    float* Q  = ws;
    float* K  = Q  + (long long)n * FDIM;
    float* V  = K  + (long long)n * FDIM;
    float* wV = V  + (long long)n * FDIM;
    float* z  = wV + (long long)n * FDIM;

    float* h_out = (float*)d_out;
    float* e_out = h_out + (long long)n * FDIM;

    // 0) zero accumulators (wV and z are contiguous)
    long long nz = (long long)n * FDIM + (long long)n * NHEAD;
    zero_kernel<<<1024, 256, 0, stream>>>(wV, nz);

    // 1) node projections
    int node_tiles  = (n + 15) / 16;
    int node_blocks = (node_tiles + 7) / 8;
    node_proj_kernel<<<node_blocks, 256, 0, stream>>>(
        node_feats, Wq, bq, Wk, bk, Wv, bv, Q, K, V, n);

    // 2) fused edge kernel
    int edge_tiles  = (e + 15) / 16;
    int edge_blocks = (edge_tiles + 7) / 8;
    edge_kernel<<<edge_blocks, 256, 0, stream>>>(
        edge_feats, src, dst, We, be, Q, K, V, e_out, wV, z, e);

    // 3) normalize
    int hn = n * FDIM;
    normalize_kernel<<<(hn + 255) / 256, 256, 0, stream>>>(wV, z, h_out, hn);
}